// FGIGScan_26491358281920
// MI455X (gfx1250) — compile-verified
//
#include <hip/hip_runtime.h>
#include <hip/hip_bf16.h>
#include <math.h>

#define BB 8
#define CCH 384
#define HH 96
#define WW 96
#define GH 24
#define GW 24
#define NG (GH*GW)        // 576 regions
#define BN_EPS 1e-5f

typedef __attribute__((ext_vector_type(2))) float v2f;
typedef __attribute__((ext_vector_type(8))) float v8f;

// fast sigmoid: v_exp_f32 + v_rcp_f32 (no IEEE divide sequence)
__device__ __forceinline__ float sigmoidf_(float v) {
    return __builtin_amdgcn_rcpf(1.0f + __expf(-v));
}

// ---------------------------------------------------------------------------
// Kernel A: fused depthwise3x3 + BN + SiLU + pointwise(C->1) via WMMA + sigmoid
// One wave (32 threads) = 16 contiguous pixels of one row.
//   lane L: pixel m = L%16, channel sub-pair half = L/16
//   A[m,k] = h(pixel m, channel 4c+k)   (K-mapping per ISA 16x4 f32 layout)
//   B[k,n] = pw_w[4c+k]  (constant across N)
//   D[m,n] = logit(pixel m)  -> identical across n; acc[v] on half h = pixel 8h+v
// Boundary handling: clamped indices + validity masks folded into the conv
// weights, so every tap load is unconditional (no exec-mask divergence).
// ---------------------------------------------------------------------------
__global__ __launch_bounds__(32)
void fg_importance_kernel(const float* __restrict__ x,
                          const float* __restrict__ dw_w,
                          const float* __restrict__ gamma,
                          const float* __restrict__ beta,
                          const float* __restrict__ mean,
                          const float* __restrict__ var,
                          const float* __restrict__ pw_w,
                          const float* __restrict__ pw_b,
                          float* __restrict__ imp)
{
    const int blk = blockIdx.x;               // B * H * (W/16)
    const int b   = blk / (HH * 6);
    const int r   = blk % (HH * 6);
    const int y   = r / 6;
    const int w0  = (r % 6) * 16;

    const int lane = threadIdx.x;
    const int m    = lane & 15;
    const int half = lane >> 4;
    const int px   = w0 + m;

    // --- uniform row clamps/masks (y is the same for the whole block) ---
    const int  yym = (y > 0)      ? y - 1 : 0;
    const int  yyp = (y < HH - 1) ? y + 1 : HH - 1;
    const float wy0 = (y > 0)      ? 1.0f : 0.0f;
    const float wy2 = (y < HH - 1) ? 1.0f : 0.0f;

    // --- per-lane column clamps/masks ---
    const int  xxm = (px > 0)      ? px - 1 : 0;
    const int  xxp = (px < WW - 1) ? px + 1 : WW - 1;
    const float wx0 = (px > 0)      ? 1.0f : 0.0f;
    const float wx2 = (px < WW - 1) ? 1.0f : 0.0f;

    // tap validity masks (multiplied into the depthwise weights)
    float msk[9];
    msk[0] = wy0 * wx0; msk[1] = wy0; msk[2] = wy0 * wx2;
    msk[3] = wx0;       msk[4] = 1.f; msk[5] = wx2;
    msk[6] = wy2 * wx0; msk[7] = wy2; msk[8] = wy2 * wx2;

    // row/column byte-free element offsets within one channel plane
    int off[9];
    off[0] = yym * WW + xxm; off[1] = yym * WW + px; off[2] = yym * WW + xxp;
    off[3] = y   * WW + xxm; off[4] = y   * WW + px; off[5] = y   * WW + xxp;
    off[6] = yyp * WW + xxm; off[7] = yyp * WW + px; off[8] = yyp * WW + xxp;

    v8f acc = {0.f, 0.f, 0.f, 0.f, 0.f, 0.f, 0.f, 0.f};

    for (int c = 0; c < CCH; c += 4) {
        const int ch0 = c + 2 * half;         // half=0 -> K=0,1 ; half=1 -> K=2,3
        const int ch1 = ch0 + 1;

        const float* p0  = x + (size_t)(b * CCH + ch0) * (HH * WW);
        const float* p1  = x + (size_t)(b * CCH + ch1) * (HH * WW);
        const float* wp0 = dw_w + ch0 * 9;
        const float* wp1 = dw_w + ch1 * 9;

        float a0 = 0.f, a1 = 0.f;
        #pragma unroll
        for (int t = 0; t < 9; ++t) {
            a0 = fmaf(p0[off[t]], wp0[t] * msk[t], a0);
            a1 = fmaf(p1[off[t]], wp1[t] * msk[t], a1);
        }

        // BatchNorm (inference) + SiLU
        const float s0 = gamma[ch0] * rsqrtf(var[ch0] + BN_EPS);
        const float s1 = gamma[ch1] * rsqrtf(var[ch1] + BN_EPS);
        float h0 = fmaf(a0, s0, beta[ch0] - mean[ch0] * s0);
        float h1 = fmaf(a1, s1, beta[ch1] - mean[ch1] * s1);
        h0 = h0 * sigmoidf_(h0);
        h1 = h1 * sigmoidf_(h1);

        v2f A;  A.x  = h0;         A.y  = h1;
        v2f Bv; Bv.x = pw_w[ch0];  Bv.y = pw_w[ch1];
        acc = __builtin_amdgcn_wmma_f32_16x16x4_f32(
                  false, A, false, Bv, (short)0, acc, false, false);
    }

    // Extract logits: D is constant across N, so lane 0 / lane 16 each hold
    // 8 finished pixels in acc[0..7].
    if (m == 0) {
        const float bias = pw_b[0];
        float* op = imp + ((size_t)b * HH + y) * WW + w0 + half * 8;
        #pragma unroll
        for (int v = 0; v < 8; ++v)
            op[v] = sigmoidf_(acc[v] + bias);
    }
}

// ---------------------------------------------------------------------------
// Kernel B: 4x4 block mean of importance -> scores[B, 576]
// ---------------------------------------------------------------------------
__global__ void fg_scores_kernel(const float* __restrict__ imp,
                                 float* __restrict__ scores)
{
    const int t = blockIdx.x * blockDim.x + threadIdx.x;
    if (t >= BB * NG) return;
    const int b  = t / NG;
    const int g  = t % NG;
    const int gh = g / GW, gw = g % GW;
    const float* p = imp + ((size_t)b * HH + gh * 4) * WW + gw * 4;
    float s = 0.f;
    #pragma unroll
    for (int rh = 0; rh < 4; ++rh)
        #pragma unroll
        for (int rw = 0; rw < 4; ++rw)
            s += p[rh * WW + rw];
    scores[t] = s * (1.0f / 16.0f);
}

// ---------------------------------------------------------------------------
// Kernel C: stable descending argsort of 576 scores per batch (rank counting)
// element i lands at rank = #{j : s[j]>s[i]} + #{j<i : s[j]==s[i]}
// ---------------------------------------------------------------------------
__global__ __launch_bounds__(NG)
void fg_argsort_kernel(const float* __restrict__ scores,
                       int* __restrict__ order)
{
    __shared__ float s[NG];
    const int b = blockIdx.x;
    const int i = threadIdx.x;
    s[i] = scores[b * NG + i];
    __syncthreads();
    const float si = s[i];
    int rank = 0;
    for (int j = 0; j < NG; ++j) {
        const float sj = s[j];
        rank += (sj > si) || (sj == si && j < i);
    }
    order[b * NG + rank] = i;
}

// ---------------------------------------------------------------------------
// Kernel D: out = region-permuted ( x * (1 + gs * importance) )
// One thread per float4 (a region row is exactly 4 floats, 16B aligned).
// ---------------------------------------------------------------------------
__global__ void fg_permute_kernel(const float* __restrict__ x,
                                  const float* __restrict__ imp,
                                  const int* __restrict__ order,
                                  const float* __restrict__ gs_p,
                                  float* __restrict__ out)
{
    const int total = BB * CCH * HH * (WW / 4);
    const int t = blockIdx.x * blockDim.x + threadIdx.x;
    if (t >= total) return;
    const float gs = gs_p[0];

    const int gw   = t % (WW / 4);            // W/4 == GW
    int rest       = t / (WW / 4);
    const int h    = rest % HH; rest /= HH;
    const int c    = rest % CCH;
    const int b    = rest / CCH;
    const int gh   = h >> 2, rh = h & 3;

    const int gdst = gh * GW + gw;
    const int gsrc = order[b * NG + gdst];
    const int sh   = (gsrc / GW) * 4 + rh;
    const int swb  = (gsrc % GW) * 4;

    const float4 xs = *reinterpret_cast<const float4*>(
        x + (((size_t)b * CCH + c) * HH + sh) * WW + swb);
    const float4 im = *reinterpret_cast<const float4*>(
        imp + ((size_t)b * HH + sh) * WW + swb);

    float4 o;
    o.x = xs.x * fmaf(gs, im.x, 1.0f);
    o.y = xs.y * fmaf(gs, im.y, 1.0f);
    o.z = xs.z * fmaf(gs, im.z, 1.0f);
    o.w = xs.w * fmaf(gs, im.w, 1.0f);

    *reinterpret_cast<float4*>(
        out + (((size_t)b * CCH + c) * HH + h) * WW + gw * 4) = o;
}

// ---------------------------------------------------------------------------
extern "C" void kernel_launch(void* const* d_in, const int* in_sizes, int n_in,
                              void* d_out, int out_size, void* d_ws, size_t ws_size,
                              hipStream_t stream) {
    const float* x     = (const float*)d_in[0];
    const float* dw_w  = (const float*)d_in[1];
    const float* gamma = (const float*)d_in[2];
    const float* beta  = (const float*)d_in[3];
    const float* mean  = (const float*)d_in[4];
    const float* var   = (const float*)d_in[5];
    const float* pw_w  = (const float*)d_in[6];
    const float* pw_b  = (const float*)d_in[7];
    const float* gs    = (const float*)d_in[8];
    float* out = (float*)d_out;

    // workspace layout
    char* ws = (char*)d_ws;
    float* imp    = (float*)ws;                                  // B*H*W floats
    float* scores = (float*)(ws + (size_t)BB * HH * WW * 4);     // B*576 floats
    int*   order  = (int*)((char*)scores + (size_t)BB * NG * 4); // B*576 ints

    // A: importance (one wave per 16 pixels)
    fg_importance_kernel<<<BB * HH * 6, 32, 0, stream>>>(
        x, dw_w, gamma, beta, mean, var, pw_w, pw_b, imp);

    // B: region scores
    {
        const int n = BB * NG;
        fg_scores_kernel<<<(n + 255) / 256, 256, 0, stream>>>(imp, scores);
    }

    // C: per-batch stable descending argsort
    fg_argsort_kernel<<<BB, NG, 0, stream>>>(scores, order);

    // D: modulate + region permute
    {
        const int n = BB * CCH * HH * (WW / 4);
        fg_permute_kernel<<<(n + 255) / 256, 256, 0, stream>>>(
            x, imp, order, gs, out);
    }
}